// Node_GAT_10505490006188
// MI455X (gfx1250) — compile-verified
//
#include <hip/hip_runtime.h>

#define N_NODES 50000
#define N_EDGES 800000
#define E_TOT   (N_EDGES + N_NODES)   // self loops appended
#define D_IN    128
#define D_HID   64
#define N_CLS   40
#define N_CLS_P 48                    // head N padded to tile multiple

typedef __attribute__((ext_vector_type(2))) float v2f;
typedef __attribute__((ext_vector_type(8))) float v8f;

// ---- order-preserving float <-> uint encoding (for atomic max on floats) ----
__device__ __forceinline__ unsigned enc_f(float f) {
  unsigned u = __float_as_uint(f);
  return (u & 0x80000000u) ? ~u : (u | 0x80000000u);
}
__device__ __forceinline__ float dec_f(unsigned u) {
  return __uint_as_float((u & 0x80000000u) ? (u & 0x7fffffffu) : ~u);
}

// ---------------------------------------------------------------------------
// C[M,Npad] = A[M,K] @ B[K,Npad] (+ bias), B pre-transposed as BT[Npad,K].
// One wave computes MT vertically-stacked 16x16 tiles (shared B fragment).
// K is a compile-time constant -> fully unrolled, guard-free MAC loop; both
// fragments load as contiguous b64. Grid: (Mtiles/MT, Npad/16), block = 32.
// Fragment maps (V_WMMA_F32_16X16X4_F32, wave32):
//   A 16x4 : lane half h, row M=l, VGPR0/1 = K = k0+2h, k0+2h+1
//   B 4x16 : lane half h, col N=l, VGPR0/1 = K = k0+2h, k0+2h+1 (via BT)
//   C      : VGPR j -> M = j + 8h, N = l
// ---------------------------------------------------------------------------
template <int K, int MT>
__global__ void gemm_wmma_f32_bt(const float* __restrict__ A,
                                 const float* __restrict__ BT,
                                 const float* __restrict__ bias,
                                 float* __restrict__ C, int ldc) {
  const int lane = threadIdx.x & 31;
  const int hi   = lane >> 4;
  const int l    = lane & 15;
  const int m0   = blockIdx.x * (16 * MT);
  const int n0   = blockIdx.y * 16;

  const float* __restrict__ bt = BT + (size_t)(n0 + l) * K;
  const float* ar[MT];
  v8f c[MT];
#pragma unroll
  for (int t = 0; t < MT; ++t) {
    ar[t] = A + (size_t)(m0 + t * 16 + l) * K;
    c[t]  = v8f{0.f, 0.f, 0.f, 0.f, 0.f, 0.f, 0.f, 0.f};
  }

#pragma unroll
  for (int k0 = 0; k0 < K; k0 += 4) {
    const int ka = k0 + hi * 2;
    const v2f b = *(const v2f*)(bt + ka);          // global_load_b64
#pragma unroll
    for (int t = 0; t < MT; ++t) {
      const v2f a = *(const v2f*)(ar[t] + ka);     // global_load_b64
      c[t] = __builtin_amdgcn_wmma_f32_16x16x4_f32(false, a, false, b,
                                                   (short)0, c[t], false, false);
    }
  }

  const float bv = (bias != nullptr) ? bias[n0 + l] : 0.f;
#pragma unroll
  for (int t = 0; t < MT; ++t) {
#pragma unroll
    for (int j = 0; j < 8; ++j) {
      C[(size_t)(m0 + t * 16 + j + hi * 8) * ldc + (n0 + l)] = c[t][j] + bv;
    }
  }
}

// BT[n*K+k] = (n < N) ? B[k*N+n] : 0   for n in [0, Npad)
__global__ void transpose_pad_kernel(const float* __restrict__ B,
                                     float* __restrict__ BT,
                                     int K, int N, int Npad) {
  const int i = blockIdx.x * blockDim.x + threadIdx.x;
  if (i >= Npad * K) return;
  const int n = i / K;
  const int k = i - n * K;
  BT[i] = (n < N) ? B[(size_t)k * N + n] : 0.f;
}

__global__ void pad_bias_kernel(const float* __restrict__ b,
                                float* __restrict__ bp, int N, int Npad) {
  const int i = blockIdx.x * blockDim.x + threadIdx.x;
  if (i >= Npad) return;
  bp[i] = (i < N) ? b[i] : 0.f;
}

// ---------------------------------------------------------------------------
// alpha_src[n] = h[n,:] . aW[0:64],  alpha_dst[n] = h[n,:] . aW[64:128]
// 8 waves per block, one wave per node; xor-shuffle reduction (wave32).
// ---------------------------------------------------------------------------
__global__ void alpha_kernel(const float* __restrict__ h,
                             const float* __restrict__ aW,
                             float* __restrict__ asrc,
                             float* __restrict__ adst) {
  const int node = blockIdx.x * 8 + (threadIdx.x >> 5);
  const int lane = threadIdx.x & 31;
  if (node >= N_NODES) return;
  const float v0 = h[(size_t)node * D_HID + lane];
  const float v1 = h[(size_t)node * D_HID + lane + 32];
  float s = v0 * aW[lane]         + v1 * aW[lane + 32];
  float t = v0 * aW[D_HID + lane] + v1 * aW[D_HID + lane + 32];
  for (int off = 16; off > 0; off >>= 1) {
    s += __shfl_xor(s, off, 32);
    t += __shfl_xor(t, off, 32);
  }
  if (lane == 0) { asrc[node] = s; adst[node] = t; }
}

__device__ __forceinline__ void edge_sd(const long long* __restrict__ ei,
                                        int e, int& s, int& d) {
  if (e < N_EDGES) {
    s = (int)ei[e];
    d = (int)ei[(size_t)N_EDGES + e];
  } else {
    s = d = e - N_EDGES;       // self loop
  }
}

// pass 1: sim = leaky_relu(alpha_src[s] + alpha_dst[d] + ab); segment max -> menc
__global__ void edge_score_kernel(const long long* __restrict__ ei,
                                  const float* __restrict__ asrc,
                                  const float* __restrict__ adst,
                                  const float* __restrict__ ab,
                                  float* __restrict__ ebuf,
                                  unsigned* __restrict__ menc) {
  const int e = blockIdx.x * blockDim.x + threadIdx.x;
  if (e >= E_TOT) return;
  int s, d; edge_sd(ei, e, s, d);
  float sim = asrc[s] + adst[d] + ab[0];
  sim = (sim > 0.f) ? sim : 0.2f * sim;
  ebuf[e] = sim;
  atomicMax(&menc[d], enc_f(sim));
}

// pass 2: ev = exp(sim - m[d]); segment sum -> z
__global__ void edge_exp_kernel(const long long* __restrict__ ei,
                                const unsigned* __restrict__ menc,
                                float* __restrict__ ebuf,   // in: sim, out: ev
                                float* __restrict__ z) {
  const int e = blockIdx.x * blockDim.x + threadIdx.x;
  if (e >= E_TOT) return;
  int s, d; edge_sd(ei, e, s, d);
  const float ev = expf(ebuf[e] - dec_f(menc[d]));
  ebuf[e] = ev;
  atomicAdd(&z[d], ev);
}

// pass 3: per (edge, feature): scatter-max (ev/z[d]) * h[s,f] into outenc[d,f]
__global__ void edge_aggr_kernel(const long long* __restrict__ ei,
                                 const float* __restrict__ ebuf,
                                 const float* __restrict__ z,
                                 const float* __restrict__ h,
                                 unsigned* __restrict__ outenc) {
  const long long t = (long long)blockIdx.x * blockDim.x + threadIdx.x;
  if (t >= (long long)E_TOT * D_HID) return;
  const int e = (int)(t >> 6);
  const int f = (int)(t & (D_HID - 1));
  int s, d; edge_sd(ei, e, s, d);
  const float a = ebuf[e] / z[d];
  const float v = a * h[(size_t)s * D_HID + f];
  atomicMax(&outenc[(size_t)d * D_HID + f], enc_f(v));
}

// decode accumulated max, add bias, relu
__global__ void finalize_kernel(const unsigned* __restrict__ accin,
                                const float* __restrict__ bias,
                                float* __restrict__ hout) {
  const int i = blockIdx.x * blockDim.x + threadIdx.x;
  if (i >= N_NODES * D_HID) return;
  const float v = dec_f(accin[i]) + bias[i & (D_HID - 1)];
  hout[i] = v > 0.f ? v : 0.f;
}

// log_softmax: read padded logits [N, 48], write d_out [N, 40]
__global__ void logsoftmax_kernel(const float* __restrict__ lp,
                                  float* __restrict__ o) {
  const int node = blockIdx.x * blockDim.x + threadIdx.x;
  if (node >= N_NODES) return;
  const float* r = lp + (size_t)node * N_CLS_P;
  float mx = r[0];
  for (int c = 1; c < N_CLS; ++c) mx = fmaxf(mx, r[c]);
  float sum = 0.f;
  for (int c = 0; c < N_CLS; ++c) sum += expf(r[c] - mx);
  const float ls = logf(sum) + mx;
  float* w = o + (size_t)node * N_CLS;
  for (int c = 0; c < N_CLS; ++c) w[c] = r[c] - ls;
}

extern "C" void kernel_launch(void* const* d_in, const int* in_sizes, int n_in,
                              void* d_out, int out_size, void* d_ws, size_t ws_size,
                              hipStream_t stream) {
  const float*     x   = (const float*)d_in[0];
  const long long* ei  = (const long long*)d_in[1];   // int64 edge_index [2,E]
  const float*     W1  = (const float*)d_in[2];
  const float*     aW1 = (const float*)d_in[3];
  const float*     ab1 = (const float*)d_in[4];
  const float*     b1  = (const float*)d_in[5];
  const float*     W2  = (const float*)d_in[6];
  const float*     aW2 = (const float*)d_in[7];
  const float*     ab2 = (const float*)d_in[8];
  const float*     b2  = (const float*)d_in[9];
  const float*     Wo  = (const float*)d_in[10];
  const float*     bo  = (const float*)d_in[11];
  float* out = (float*)d_out;

  // workspace carve-up (~40 MB)
  char* w = (char*)d_ws;
  float*    bufA  = (float*)w;    w += (size_t)N_NODES * D_HID * sizeof(float);  // h pre-aggr
  float*    bufB  = (float*)w;    w += (size_t)N_NODES * D_HID * sizeof(float);  // acc / h post
  float*    lgts  = (float*)w;    w += (size_t)N_NODES * N_CLS_P * sizeof(float);
  float*    asrc  = (float*)w;    w += (size_t)N_NODES * sizeof(float);
  float*    adst  = (float*)w;    w += (size_t)N_NODES * sizeof(float);
  unsigned* menc  = (unsigned*)w; w += (size_t)N_NODES * sizeof(unsigned);
  float*    z     = (float*)w;    w += (size_t)N_NODES * sizeof(float);
  float*    ebuf  = (float*)w;    w += (size_t)E_TOT * sizeof(float);
  float*    bt1   = (float*)w;    w += (size_t)D_HID * D_IN * sizeof(float);     // W1^T
  float*    bt2   = (float*)w;    w += (size_t)D_HID * D_HID * sizeof(float);    // W2^T
  float*    bto   = (float*)w;    w += (size_t)N_CLS_P * D_HID * sizeof(float);  // Wo^T pad
  float*    bop   = (float*)w;    w += (size_t)N_CLS_P * sizeof(float);          // bo pad

  const dim3 wave(32);
  const int eb  = (E_TOT + 255) / 256;
  const int ab3 = (int)(((long long)E_TOT * D_HID + 255) / 256);
  const int nfb = (N_NODES * D_HID + 255) / 256;
  const int MT  = 5;                       // 3125 M-tiles = 625 * 5
  const dim3 gemm_grid(N_NODES / (16 * MT), D_HID / 16);
  const dim3 head_grid(N_NODES / (16 * MT), N_CLS_P / 16);

  // ---- weight prep (transpose + pad; tiny) ----
  transpose_pad_kernel<<<(D_HID * D_IN + 255) / 256, 256, 0, stream>>>(W1, bt1, D_IN, D_HID, D_HID);
  transpose_pad_kernel<<<(D_HID * D_HID + 255) / 256, 256, 0, stream>>>(W2, bt2, D_HID, D_HID, D_HID);
  transpose_pad_kernel<<<(N_CLS_P * D_HID + 255) / 256, 256, 0, stream>>>(Wo, bto, D_HID, N_CLS, N_CLS_P);
  pad_bias_kernel<<<1, 64, 0, stream>>>(bo, bop, N_CLS, N_CLS_P);

  // ================= Layer 1 =================
  gemm_wmma_f32_bt<D_IN, 5><<<gemm_grid, wave, 0, stream>>>(x, bt1, nullptr, bufA, D_HID);
  alpha_kernel<<<(N_NODES + 7) / 8, 256, 0, stream>>>(bufA, aW1, asrc, adst);
  hipMemsetAsync(menc, 0, (size_t)N_NODES * 4, stream);
  hipMemsetAsync(z,    0, (size_t)N_NODES * 4, stream);
  hipMemsetAsync(bufB, 0, (size_t)N_NODES * D_HID * 4, stream);
  edge_score_kernel<<<eb, 256, 0, stream>>>(ei, asrc, adst, ab1, ebuf, menc);
  edge_exp_kernel  <<<eb, 256, 0, stream>>>(ei, menc, ebuf, z);
  edge_aggr_kernel <<<ab3, 256, 0, stream>>>(ei, ebuf, z, bufA, (unsigned*)bufB);
  finalize_kernel  <<<nfb, 256, 0, stream>>>((const unsigned*)bufB, b1, bufB);

  // ================= Layer 2 =================
  gemm_wmma_f32_bt<D_HID, 5><<<gemm_grid, wave, 0, stream>>>(bufB, bt2, nullptr, bufA, D_HID);
  alpha_kernel<<<(N_NODES + 7) / 8, 256, 0, stream>>>(bufA, aW2, asrc, adst);
  hipMemsetAsync(menc, 0, (size_t)N_NODES * 4, stream);
  hipMemsetAsync(z,    0, (size_t)N_NODES * 4, stream);
  hipMemsetAsync(bufB, 0, (size_t)N_NODES * D_HID * 4, stream);
  edge_score_kernel<<<eb, 256, 0, stream>>>(ei, asrc, adst, ab2, ebuf, menc);
  edge_exp_kernel  <<<eb, 256, 0, stream>>>(ei, menc, ebuf, z);
  edge_aggr_kernel <<<ab3, 256, 0, stream>>>(ei, ebuf, z, bufA, (unsigned*)bufB);
  finalize_kernel  <<<nfb, 256, 0, stream>>>((const unsigned*)bufB, b2, bufB);

  // ================= Output head + log_softmax =================
  gemm_wmma_f32_bt<D_HID, 5><<<head_grid, wave, 0, stream>>>(bufB, bto, bop, lgts, N_CLS_P);
  logsoftmax_kernel<<<(N_NODES + 255) / 256, 256, 0, stream>>>(lgts, out);
}